// _SpanMasker_36996848288163
// MI455X (gfx1250) — compile-verified
//
#include <hip/hip_runtime.h>
#include <hip/hip_bf16.h>
#include <stdint.h>

// ---------------------------------------------------------------------------
// SpanMasker for MI455X (gfx1250).
// Phase A: per-row run detection + random run selection (tiny, LDS-resident).
// Phase B: HBM-streaming mask apply. ~201 MB total traffic -> ~8.7us at
// 23.3 TB/s. Pure bandwidth op: double-buffered CDNA5 async global->LDS
// copies (ASYNCcnt) pipelined against scale + nontemporal B128 stores.
// ---------------------------------------------------------------------------

#define TFRAMES 2048
#define DDIM    768
#define BLK     256
#define PER     (TFRAMES / BLK)   // 8 frames per thread in the scan
#define NTILES  4                 // tiles per block in the streaming kernel

typedef float v4f __attribute__((ext_vector_type(4)));
typedef int   v4i __attribute__((ext_vector_type(4)));

typedef __attribute__((address_space(1))) v4i* gptr_v4i;   // global
typedef __attribute__((address_space(3))) v4i* lptr_v4i;   // LDS

#if defined(__has_builtin)
#  if __has_builtin(__builtin_amdgcn_global_load_async_to_lds_b128)
#    define HAVE_ASYNC_LDS 1
#  endif
#  if __has_builtin(__builtin_amdgcn_s_wait_asynccnt)
#    define HAVE_WAIT_ASYNC 1
#  endif
#endif
#ifndef HAVE_ASYNC_LDS
#  define HAVE_ASYNC_LDS 0
#endif
#ifndef HAVE_WAIT_ASYNC
#  define HAVE_WAIT_ASYNC 0
#endif

template <int N>
__device__ __forceinline__ void wait_async() {
#if HAVE_WAIT_ASYNC
    __builtin_amdgcn_s_wait_asynccnt(N);
#else
    asm volatile("s_wait_asynccnt %0" :: "n"(N) : "memory");
#endif
    asm volatile("" ::: "memory");   // LDS reads must observe the DMA
}

__device__ __forceinline__ unsigned hash31(unsigned x) {
    // splitmix32 finalizer; >>1 keeps result < 0xFFFFFFFF (sentinel-safe)
    x ^= x >> 16; x *= 0x7feb352dU;
    x ^= x >> 15; x *= 0x846ca68bU;
    x ^= x >> 16;
    return x >> 1;
}

// -------------------- Phase A: per-row span mask ---------------------------
// One workgroup (8 waves) per row. All state lives in LDS (~37 KB of the
// 320 KB/WGP pool).
__global__ void __launch_bounds__(BLK)
span_mask_kernel(const int* __restrict__ ids, float* __restrict__ maskOut) {
    __shared__ int      sIds[TFRAMES];
    __shared__ int      sRunIdx[TFRAMES];
    __shared__ int      sRunId[TFRAMES];
    __shared__ unsigned sScore[TFRAMES];
    __shared__ unsigned char sChosen[TFRAMES];
    __shared__ int      sPart[BLK];
    __shared__ int      sCnt[BLK];
    __shared__ int      sK;

    const int row = blockIdx.x;
    const int tid = threadIdx.x;
    const int* idrow = ids + (size_t)row * TFRAMES;

    for (int t = tid; t < TFRAMES; t += BLK) sIds[t] = idrow[t];
    __syncthreads();

    // boundary flags + thread-local inclusive scan (8 frames/thread)
    const int base = tid * PER;
    int loc[PER];
    int acc = 0;
    #pragma unroll
    for (int j = 0; j < PER; ++j) {
        const int t = base + j;
        const int b = (t == 0) ? 1 : (sIds[t] != sIds[t - 1]);
        acc += b;
        loc[j] = acc;
    }
    sPart[tid] = acc;
    __syncthreads();

    // Hillis-Steele scan of the 256 partials
    for (int off = 1; off < BLK; off <<= 1) {
        const int mine = sPart[tid];
        const int add  = (tid >= off) ? sPart[tid - off] : 0;
        __syncthreads();
        sPart[tid] = mine + add;
        __syncthreads();
    }
    const int prev = (tid > 0) ? sPart[tid - 1] : 0;
    #pragma unroll
    for (int j = 0; j < PER; ++j)
        sRunIdx[base + j] = prev + loc[j] - 1;   // run slot per frame
    __syncthreads();

    const int nRuns = sPart[BLK - 1];

    // scatter each run's id from its first frame
    for (int t = tid; t < TFRAMES; t += BLK) {
        const int b = (t == 0) ? 1 : (sIds[t] != sIds[t - 1]);
        if (b) sRunId[sRunIdx[t]] = sIds[t];
    }
    __syncthreads();

    // candidates (non-zero id) get a random score; others get the sentinel
    int cnt = 0;
    for (int s = tid; s < nRuns; s += BLK) {
        const bool cand = (sRunId[s] != 0);
        cnt += cand ? 1 : 0;
        sScore[s] = cand ? hash31(0x9e3779b9U ^ (unsigned)(row * TFRAMES + s))
                         : 0xFFFFFFFFu;
    }
    sCnt[tid] = cnt;
    __syncthreads();
    for (int off = BLK / 2; off > 0; off >>= 1) {
        if (tid < off) sCnt[tid] += sCnt[tid + off];
        __syncthreads();
    }
    if (tid == 0) {
        const int nCand = sCnt[0];
        int k = (int)rintf(0.2f * (float)nCand);  // round-half-even like jnp.round
        sK = (k < 1) ? 1 : k;
    }
    __syncthreads();
    const int k = sK;

    // rank-by-count selection of the k lowest-score candidates (R<=2048)
    for (int s = tid; s < nRuns; s += BLK) {
        const unsigned my = sScore[s];
        unsigned char ch = 0;
        if (my != 0xFFFFFFFFu) {
            int r = 0;
            for (int s2 = 0; s2 < nRuns; ++s2) {
                const unsigned u = sScore[s2];
                r += (u < my) || (u == my && s2 < s);
            }
            ch = (r < k) ? 1 : 0;
        }
        sChosen[s] = ch;
    }
    __syncthreads();

    // broadcast run decision back to frames; mask stored as 0.0/1.0 float
    float* mrow = maskOut + (size_t)row * TFRAMES;
    for (int t = tid; t < TFRAMES; t += BLK)
        mrow[t] = (float)sChosen[sRunIdx[t]];
}

// -------------------- Phase B: streaming mask apply ------------------------
// Each block streams NTILES x 4KB tiles. Async DMA into a double-buffered
// LDS tile overlaps with scale + nontemporal store of the previous tile:
//   issue(t+1) -> s_wait_asynccnt 1 -> consume(t) -> store NT
__global__ void __launch_bounds__(BLK)
apply_mask_kernel(const float* __restrict__ x,
                  const float* __restrict__ maskv,
                  float* __restrict__ out, int n4) {
#if HAVE_ASYNC_LDS
    __shared__ v4f tile[2][BLK];
#endif
    const int tid  = threadIdx.x;
    const int base = blockIdx.x * (BLK * NTILES) + tid;
    const v4f* x4  = (const v4f*)x;
    v4f* o4        = (v4f*)out;

#if HAVE_ASYNC_LDS
    // prologue: start tile 0's DMA
    if (base < n4)
        __builtin_amdgcn_global_load_async_to_lds_b128(
            (gptr_v4i)(x4 + base), (lptr_v4i)&tile[0][tid], 0, 0);

    #pragma unroll
    for (int t = 0; t < NTILES; ++t) {
        const int i   = base + t * BLK;
        const int nxt = base + (t + 1) * BLK;

        // keep one DMA in flight into the other buffer
        if (t + 1 < NTILES && nxt < n4)
            __builtin_amdgcn_global_load_async_to_lds_b128(
                (gptr_v4i)(x4 + nxt), (lptr_v4i)&tile[(t + 1) & 1][tid], 0, 0);

        // speculative look-ahead into L2, two tiles out
        const int pf = base + (t + 2) * BLK;
        if (pf < n4) __builtin_prefetch(x4 + pf, 0, 0);

        if (t + 1 < NTILES) wait_async<1>();   // tile t done, t+1 in flight
        else                wait_async<0>();   // drain

        if (i < n4) {
            v4f v = tile[t & 1][tid];
            const float m = maskv[i / (DDIM / 4)];   // 192 float4 per frame
            v *= (m != 0.0f) ? 0.0f : 1.0f;
            __builtin_nontemporal_store(v, o4 + i);
        }
    }
#else
    #pragma unroll
    for (int t = 0; t < NTILES; ++t) {
        const int i = base + t * BLK;
        if (i < n4) {
            v4f v = __builtin_nontemporal_load(x4 + i);
            const float m = maskv[i / (DDIM / 4)];
            v *= (m != 0.0f) ? 0.0f : 1.0f;
            __builtin_nontemporal_store(v, o4 + i);
        }
    }
#endif
}

// ---------------------------------------------------------------------------
extern "C" void kernel_launch(void* const* d_in, const int* in_sizes, int n_in,
                              void* d_out, int out_size, void* d_ws, size_t ws_size,
                              hipStream_t stream) {
    const float* x   = (const float*)d_in[0];
    const int*   ids = (const int*)d_in[1];

    const int BT = in_sizes[1];          // B*T
    const int B  = BT / TFRAMES;

    float* out     = (float*)d_out;
    float* maskOut = out + (size_t)BT * DDIM;   // mask tail of d_out

    // Phase A: one workgroup per row
    span_mask_kernel<<<B, BLK, 0, stream>>>(ids, maskOut);

    // Phase B: stream x through double-buffered async LDS tiles
    const int n4     = BT * (DDIM / 4);
    const int blocks = (n4 + BLK * NTILES - 1) / (BLK * NTILES);
    apply_mask_kernel<<<blocks, BLK, 0, stream>>>(x, maskOut, out, n4);
}